// LiftingScheme_39032662786092
// MI455X (gfx1250) — compile-verified
//
#include <hip/hip_runtime.h>

typedef __attribute__((ext_vector_type(2))) float v2f;
typedef __attribute__((ext_vector_type(8))) float v8f;
typedef __attribute__((ext_vector_type(4))) int   v4i;

#define CIN   32
#define COUT  64
#define HH    56
#define WWID  56
#define HW    (HH * WWID)      // 3136
#define KTOT  (9 * CIN)        // 288
#define WBLK  (16 * CIN * 9)   // 4608 floats: raw 3x3 weights for one o-tile

#if __has_builtin(__builtin_amdgcn_global_load_async_to_lds_b128)
#define HAVE_ASYNC_LDS 1
#pragma message("CDNA5: using global_load_async_to_lds_b128 path")
#else
#define HAVE_ASYNC_LDS 0
#pragma message("CDNA5: async-to-LDS builtin NOT available, using sync fallback")
#endif

// Fused lifting-scheme conv:
//  1) async-DMA the o-tile's raw weights (contiguous 18 KB) into LDS
//  2) compute effective 3x3 weights (lifting algebra) into an A-panel in LDS
//  3) implicit GEMM via V_WMMA_F32_16X16X4_F32:
//     M = o-tile (16), N = 16 flat pixels per wave, K = 9*C = 288 (tap-major)
__global__ __launch_bounds__(256)
void lifting_conv_wmma(const float* __restrict__ x,
                       const float* __restrict__ wgt,
                       float* __restrict__ out) {
    __shared__ float Wraw[WBLK];        // 18 KB raw weights (async staged)
    __shared__ float Alds[KTOT * 16];   // 18 KB A-panel [k = t*32+c][m]

    const int tid = threadIdx.x;
    const int ot  = blockIdx.x & 3;     // o-tile (0..3)
    const int pb  = blockIdx.x >> 2;    // pixel block (0..48)

    // ---- stage raw weights for this o-tile into LDS ----
    const float* wsrc = wgt + ot * WBLK;
#if HAVE_ASYNC_LDS
    for (int q = tid; q < WBLK / 4; q += 256) {     // 1152 x b128 transfers
        __builtin_amdgcn_global_load_async_to_lds_b128(
            (v4i*)(wsrc + q * 4), (v4i*)(Wraw + q * 4), 0, 0);
    }
#if __has_builtin(__builtin_amdgcn_s_wait_asynccnt)
    __builtin_amdgcn_s_wait_asynccnt(0);
#else
    asm volatile("s_wait_asynccnt 0x0" ::: "memory");
#endif
#else
    for (int q = tid; q < WBLK / 4; q += 256) {
        const float4 v = ((const float4*)wsrc)[q];
        ((float4*)Wraw)[q] = v;
    }
#endif
    __syncthreads();

    // ---- lifting algebra: raw w (LDS) -> effective 3x3 A-panel (LDS) ----
    for (int idx = tid; idx < 16 * CIN; idx += 256) {
        const int ol = idx >> 5;        // o_local 0..15
        const int c  = idx & 31;
        const float* wp = Wraw + (ol * CIN + c) * 9;
        const float w0 = wp[0], w1 = wp[1], w2 = wp[2];
        const float w3 = wp[3], w4 = wp[4], w5 = wp[5];
        const float w6 = wp[6], w7 = wp[7], w8 = wp[8];
        const float p0 = -w6 / w7;
        const float p1 = -w2 / w5;
        const float p2 = (1.0f - w0 + w6 * w1 / w7 + w2 * w3 / w5) / w4;
        const float p3 = -w8 / w4;
        const float u0 = w1, u1 = w7, u2 = w3, u3 = w5, u4 = w4;
        float eff[9];
        eff[0] = 1.0f - u0 * p0 - u2 * p1 - u4 * p2;
        eff[1] = u0;
        eff[2] = -u3 * p1;
        eff[3] = u2;
        eff[4] = u4;
        eff[5] = u3;
        eff[6] = -u1 * p0;
        eff[7] = u1;
        eff[8] = -u4 * p3;
#pragma unroll
        for (int t = 0; t < 9; ++t)
            Alds[(t * CIN + c) * 16 + ol] = eff[t];
    }
    __syncthreads();

    // ---- implicit GEMM: one 16(o) x 16(pixel) tile per wave, K = 288 ----
    const int lane = tid & 31;
    const int wv   = tid >> 5;          // wave 0..7
    const int half = lane >> 4;         // 0|1 : K pair / D row half
    const int nl   = lane & 15;         // m for A-frag, n (pixel) for B-frag

    const int pix = (pb * 8 + wv) * 16 + nl;     // < 6272; tiles never span images
    const int img = (pix >= HW) ? 1 : 0;
    const int rem = pix - img * HW;
    const int h   = rem / WWID;
    const int w   = rem - h * WWID;

    v8f acc = {};

    for (int t = 0; t < 9; ++t) {                // tap loop: (dh,dw) fixed per t
        const int dh = t / 3 - 1;
        const int dw = t % 3 - 1;
        const int hh = h + dh;
        const int ww = w + dw;
        const bool valid = (hh >= 0) & (hh < HH) & (ww >= 0) & (ww < WWID);
        // base for c = 2*half at this tap (never dereferenced when !valid)
        const float* bbase = x + ((img * CIN + 2 * half) * HH + hh) * WWID + ww;
        const float* abase = Alds + (t * CIN + 2 * half) * 16 + nl;
        if (valid) __builtin_prefetch(bbase, 0, 1);   // global_prefetch_b8

#pragma unroll
        for (int cs = 0; cs < 8; ++cs) {         // c = cs*4 + 2*half + j
            v2f a, b;
            a.x = abase[(cs * 4) * 16];
            a.y = abase[(cs * 4 + 1) * 16];
            if (valid) {
                b.x = bbase[(cs * 4) * HW];
                b.y = bbase[(cs * 4) * HW + HW];
            } else {
                b.x = 0.0f;
                b.y = 0.0f;
            }
            // D = A(16x4,f32) * B(4x16,f32) + C  -> v_wmma_f32_16x16x4_f32
            acc = __builtin_amdgcn_wmma_f32_16x16x4_f32(
                false, a, false, b, (short)0, acc, false, false);
        }
    }

    // ---- write D: VGPR v holds row M = v + 8*half, col N = nl ----
    float* obase = out + ((img * COUT + ot * 16 + 8 * half) * HH + h) * WWID + w;
#pragma unroll
    for (int v = 0; v < 8; ++v)
        obase[v * HW] = acc[v];
}

extern "C" void kernel_launch(void* const* d_in, const int* in_sizes, int n_in,
                              void* d_out, int out_size, void* d_ws, size_t ws_size,
                              hipStream_t stream) {
    const float* x   = (const float*)d_in[0];
    const float* wgt = (const float*)d_in[1];
    float* out       = (float*)d_out;
    // 4 o-tiles x 49 pixel-blocks; 256 threads = 8 wave32 per block
    lifting_conv_wmma<<<dim3(196), dim3(256), 0, stream>>>(x, wgt, out);
}